// BatchBlobLoss_80882824118629
// MI455X (gfx1250) — compile-verified
//
#include <hip/hip_runtime.h>

// BatchBlobLoss on MI455X (gfx1250): bandwidth-bound streaming histogram.
// ~168 MB of single-pass traffic -> ~7.2 us roofline at 23.3 TB/s.
// Core: softmax (exp/rcp) + 33-bin LDS-atomic segment-sum with 8-way
// replication; targets stream staged through LDS via the CDNA5 async
// engine (ASYNCcnt) when the builtin is available.

#define NSEG  33
#define EPSF  1e-5f
#define NREP  8
#define BB    2
#define CC    3

typedef float v4f __attribute__((ext_vector_type(4)));
typedef int   v4i __attribute__((ext_vector_type(4)));

#if defined(__has_builtin)
#  if __has_builtin(__builtin_amdgcn_global_load_async_to_lds_b128) && \
      __has_builtin(__builtin_amdgcn_s_wait_asynccnt)
#    define HAVE_ASYNC_LDS 1
#  endif
#endif
#ifndef HAVE_ASYNC_LDS
#  define HAVE_ASYNC_LDS 0
#endif

#if HAVE_ASYNC_LDS
// global_load_async_to_lds_b128: (AS1 v4i*, AS3 v4i*, imm offset, imm cpol)
__device__ __forceinline__ void async_cp16(const v4i* g, v4i* l) {
  __builtin_amdgcn_global_load_async_to_lds_b128(
      (__attribute__((address_space(1))) v4i*)(g),
      (__attribute__((address_space(3))) v4i*)(l), 0, 0);
}
#endif

__global__ __launch_bounds__(256) void blob_init(float* gps, unsigned* gct) {
  int t = threadIdx.x;
  if (t < (BB * (CC - 1)) * NSEG) { gps[t] = 0.0f; gct[t] = 0u; }
}

__global__ __launch_bounds__(256) void blob_hist(
    const float* __restrict__ pred, const int* __restrict__ targ,
    float* __restrict__ gps, unsigned* __restrict__ gct, int S) {
  const int Nv  = S >> 2;              // vec4 elements per (b, channel) plane
  const int b   = blockIdx.y;
  const int tid = threadIdx.x;
  const int rep = tid & (NREP - 1);

  __shared__ float    hp[NREP][2][NSEG];
  __shared__ unsigned hc[NREP][2][NSEG];
#if HAVE_ASYNC_LDS
  __shared__ v4i tbuf[2][2][256];      // [parity][stream t1/t2][tid]
#endif

  for (int i = tid; i < NREP * 2 * NSEG; i += blockDim.x) {
    ((float*)hp)[i]    = 0.0f;
    ((unsigned*)hc)[i] = 0u;
  }
  __syncthreads();

  const size_t base = (size_t)b * CC * (size_t)S;
  const v4f* p0 = (const v4f*)(pred + base);
  const v4f* p1 = (const v4f*)(pred + base + (size_t)S);
  const v4f* p2 = (const v4f*)(pred + base + 2 * (size_t)S);
  const v4i* t1 = (const v4i*)(targ + base + (size_t)S);
  const v4i* t2 = (const v4i*)(targ + base + 2 * (size_t)S);

  const int stride = gridDim.x * blockDim.x;
  const int i0     = blockIdx.x * blockDim.x + tid;

#if HAVE_ASYNC_LDS
  int par = 0;
  if (i0 < Nv) {
    async_cp16(t1 + i0, &tbuf[0][0][tid]);
    async_cp16(t2 + i0, &tbuf[0][1][tid]);
  }
#endif

  for (int i = i0; i < Nv; i += stride) {
    v4f x0 = __builtin_nontemporal_load(p0 + i);
    v4f x1 = __builtin_nontemporal_load(p1 + i);
    v4f x2 = __builtin_nontemporal_load(p2 + i);

    v4i tv1, tv2;
#if HAVE_ASYNC_LDS
    const int next = i + stride;
    if (next < Nv) {
      async_cp16(t1 + next, &tbuf[par ^ 1][0][tid]);
      async_cp16(t2 + next, &tbuf[par ^ 1][1][tid]);
      __builtin_amdgcn_s_wait_asynccnt(2);   // oldest pair (this iter) landed
    } else {
      __builtin_amdgcn_s_wait_asynccnt(0);   // drain for the final iteration
    }
    __asm__ __volatile__("" ::: "memory");   // LDS now holds fresh targets
    tv1 = tbuf[par][0][tid];
    tv2 = tbuf[par][1][tid];
    par ^= 1;
#else
    tv1 = __builtin_nontemporal_load(t1 + i);
    tv2 = __builtin_nontemporal_load(t2 + i);
#endif

#pragma unroll
    for (int e = 0; e < 4; ++e) {
      float e0 = __expf(x0[e]);
      float e1 = __expf(x1[e]);
      float e2 = __expf(x2[e]);
      float r  = __builtin_amdgcn_rcpf(e0 + e1 + e2);
      float q1 = e1 * r;
      float q2 = e2 * r;
      int   u1 = tv1[e];
      int   u2 = tv2[e];
      atomicAdd(&hp[rep][0][u1], q1);
      atomicAdd(&hc[rep][0][u1], 1u);
      atomicAdd(&hp[rep][1][u2], q2);
      atomicAdd(&hc[rep][1][u2], 1u);
    }
  }

  __syncthreads();
  for (int idx = tid; idx < 2 * NSEG; idx += blockDim.x) {
    const int r = (idx >= NSEG) ? 1 : 0;
    const int j = idx - r * NSEG;
    float    s = 0.0f;
    unsigned c = 0u;
#pragma unroll
    for (int k = 0; k < NREP; ++k) { s += hp[k][r][j]; c += hc[k][r][j]; }
    atomicAdd(&gps[(b * 2 + r) * NSEG + j], s);
    atomicAdd(&gct[(b * 2 + r) * NSEG + j], c);
  }
}

__global__ __launch_bounds__(128) void blob_final(
    const float* __restrict__ gps, const unsigned* __restrict__ gct,
    float* __restrict__ out) {
  __shared__ float num[2], den[2];
  const int t = threadIdx.x;                 // 0..127
  if (t < 2) { num[t] = 0.0f; den[t] = 0.0f; }
  __syncthreads();

  const int b   = t >> 6;
  const int cm  = (t >> 5) & 1;
  const int i   = (t & 31) + 1;              // bins 1..32
  const int row = b * 2 + cm;

  const float sbg  = gps[row * NSEG + 0];
  const float si   = gps[row * NSEG + i];
  const float ni   = (float)gct[row * NSEG + i];
  const float dice = 1.0f - (2.0f * si + EPSF) / (sbg + si + ni + EPSF);
  const float pres = (ni > 0.0f) ? 1.0f : 0.0f;

  atomicAdd(&num[cm], dice * pres);
  atomicAdd(&den[cm], pres);
  __syncthreads();

  if (t == 0) {
    const float pc0 = num[0] / fmaxf(den[0], 1.0f);
    const float pc1 = num[1] / fmaxf(den[1], 1.0f);
    out[0] = 0.5f * (pc0 + pc1);
  }
}

extern "C" void kernel_launch(void* const* d_in, const int* in_sizes, int n_in,
                              void* d_out, int out_size, void* d_ws, size_t ws_size,
                              hipStream_t stream) {
  const float* pred = (const float*)d_in[0];
  const int*   targ = (const int*)d_in[1];

  float*    gps = (float*)d_ws;                                        // 132 floats
  unsigned* gct = (unsigned*)((char*)d_ws + 4 * NSEG * sizeof(float)); // 132 u32

  const int S = in_sizes[0] / (BB * CC);   // 64*256*256 = 4,194,304

  blob_init<<<1, 256, 0, stream>>>(gps, gct);

  dim3 grid(1024, BB);                     // stride 262144; exactly 4 uniform iters
  blob_hist<<<grid, 256, 0, stream>>>(pred, targ, gps, gct, S);

  blob_final<<<1, 128, 0, stream>>>(gps, gct, (float*)d_out);
}